// MeshNN_23330262352119
// MI455X (gfx1250) — compile-verified
//
#include <hip/hip_runtime.h>
#include <hip/hip_bf16.h>

typedef __attribute__((ext_vector_type(2))) float v2f;
typedef __attribute__((ext_vector_type(8))) float v8f;

#define NP_NODES 512

// Hat-basis interpolation. Per wave iteration: 32 points are evaluated
// (two hat terms t0,t1 per point, striped across lanes as the B matrix of a
// 16x16x4 f32 WMMA). A constant 0/1 selector matrix A makes
// D[0][n] = t0(p_n)+t1(p_n) for lanes 0..15 and D[8][n] for lanes 16..31,
// so D VGPR0, lane L == u(point base+L): one coalesced store per lane.
__global__ __launch_bounds__(256)
void meshnn_hat_wmma_kernel(const float* __restrict__ x,
                            const float* __restrict__ coords,
                            const float* __restrict__ w_uu,
                            const float* __restrict__ w_dd,
                            float* __restrict__ out,
                            int n)
{
    __shared__ float sc[NP_NODES]; // node coordinates
    __shared__ float wf[NP_NODES]; // merged weights: [w_dd0, w_uu..., w_dd1]

    for (int j = threadIdx.x; j < NP_NODES; j += blockDim.x) {
        sc[j] = coords[j];
        float w;
        if (j == 0)                w = w_dd[0];
        else if (j == NP_NODES-1)  w = w_dd[1];
        else                       w = w_uu[j - 1];
        wf[j] = w;
    }
    __syncthreads();

    const int lane   = threadIdx.x & 31;
    const int idx0   = blockIdx.x * blockDim.x + threadIdx.x; // lane-contiguous
    const int stride = gridDim.x * blockDim.x;

    const float c0   = sc[0];
    const float invh = (float)(NP_NODES - 1) / (sc[NP_NODES - 1] - c0);

    // Selector matrix A (16x4 f32, striped across VGPRs within a lane):
    // lane L<16 holds row m=L, K=0..1; lane L>=16 holds row m=L-16, K=2..3.
    // A[m][k] = 1 iff (m<8)==(k<2)  =>  per-lane value:
    const float asel = ((lane < 8) || (lane >= 24)) ? 1.0f : 0.0f;
    const v2f a = { asel, asel };
    const v8f czero = {};

    for (int idx = idx0; (idx - lane) < n; idx += stride) {
        // uniform-guarded prefetch of the next stride (global_prefetch_b8)
        if ((idx - lane) + stride < n)
            __builtin_prefetch(x + idx + stride, 0, 0);

        const int ld = (idx < n) ? idx : (n - 1);
        const float xv = x[ld];

        // locate interval [sc[i], sc[i+1]] containing xv
        int i = (int)((xv - c0) * invh);
        i = (i < 0) ? 0 : ((i > NP_NODES - 2) ? NP_NODES - 2 : i);
        while (i > 0 && xv < sc[i]) --i;
        while (i < NP_NODES - 2 && xv >= sc[i + 1]) ++i;

        const float cL  = sc[i];
        const float cR  = sc[i + 1];
        // v_rcp_f32 (<=1 ulp) instead of the full IEEE divide sequence
        const float inv = __builtin_amdgcn_rcpf(cR - cL);
        // right branch of hat_i and left branch of hat_{i+1} (reference forms)
        const float dR = fmaxf(0.0f, 1.0f - (xv - cL) * inv);
        const float dL = fmaxf(0.0f, 1.0f - (cR - xv) * inv);
        const float t0 = dR * wf[i];
        const float t1 = dL * wf[i + 1];

        // B matrix (4x16 f32): VGPR0 = rows K=0 (lanes 0-15) / K=2 (16-31),
        // VGPR1 = rows K=1 / K=3. Our per-lane (t0,t1) lands exactly there.
        const v2f b = { t0, t1 };

        // D = A x B + 0 : emits v_wmma_f32_16x16x4_f32
        v8f d = __builtin_amdgcn_wmma_f32_16x16x4_f32(
            /*neg_a=*/false, a, /*neg_b=*/false, b,
            /*c_mod=*/(short)0, czero, /*reuse_a=*/false, /*reuse_b=*/false);

        if (idx < n) out[idx] = d[0];
    }
}

extern "C" void kernel_launch(void* const* d_in, const int* in_sizes, int n_in,
                              void* d_out, int out_size, void* d_ws, size_t ws_size,
                              hipStream_t stream)
{
    const float* x      = (const float*)d_in[0]; // (N,1) f32
    const float* coords = (const float*)d_in[1]; // (512,) f32
    const float* w_uu   = (const float*)d_in[2]; // (510,) f32
    const float* w_dd   = (const float*)d_in[3]; // (2,)   f32
    float* out          = (float*)d_out;         // (N,1)  f32

    const int n = in_sizes[0];
    const int threads = 256;
    const int blocks  = 256; // 2048 wave32; grid-stride covers N=262144 in 4 steps

    meshnn_hat_wmma_kernel<<<blocks, threads, 0, stream>>>(
        x, coords, w_uu, w_dd, out, n);
}